// VectorQuantizer_57655640981630
// MI455X (gfx1250) — compile-verified
//
#include <hip/hip_runtime.h>
#include <math.h>

// ---- problem constants (match reference) ----
#define NUM_CODES 1024
#define D_DIM     256
#define N_ROWS    32768            // B*T = 32*1024
#define OUT_Q_ELEMS (N_ROWS * D_DIM)

typedef __attribute__((ext_vector_type(2))) float v2f;
typedef __attribute__((ext_vector_type(8))) float v8f;

__device__ __forceinline__ void atomicAddF(float* p, float v) {
  // HW float atomic (global_atomic_add_f32) instead of CAS loop
  unsafeAtomicAdd(p, v);
}

// ---------------------------------------------------------------------------
// Kernel 1: codebook row norms ||e_k||^2 ; also zero counts / sse scratch.
// 1024 blocks x 256 threads, one block per code row (coalesced).
// ---------------------------------------------------------------------------
__global__ __launch_bounds__(256) void k_norms(const float* __restrict__ cb,
                                               float* __restrict__ norms,
                                               float* __restrict__ counts,
                                               float* __restrict__ sse) {
  __shared__ float red[8];
  const int k = blockIdx.x;
  const int t = threadIdx.x;
  float v = cb[k * D_DIM + t];
  float s = v * v;
#pragma unroll
  for (int off = 16; off >= 1; off >>= 1) s += __shfl_xor(s, off, 32);
  if ((t & 31) == 0) red[t >> 5] = s;
  __syncthreads();
  if (t == 0) {
    float tot = 0.f;
#pragma unroll
    for (int i = 0; i < 8; ++i) tot += red[i];
    norms[k]  = tot;
    counts[k] = 0.f;
    if (k == 0) *sse = 0.f;
  }
}

// ---------------------------------------------------------------------------
// Kernel 2: fused distance-matmul + argmin via V_WMMA_F32_16X16X4_F32.
// Block = 128 threads (4 waves). Block tile = 64 rows of x staged in LDS
// (row stride 258 floats -> conflict-free A-fragment ds_load_b64).
// Each wave computes a 16-row x 16-code WMMA tile per column-tile, K=256 in
// steps of 4, and tracks min(||e||^2 - 2 z.e) per row.
// ---------------------------------------------------------------------------
#define TILE_ROWS 64
#define XS_STRIDE 258   // 258 % 64 == 2 -> 16 consecutive rows hit 16 banks

__global__ __launch_bounds__(128) void k_vq(const float* __restrict__ x,
                                            const float* __restrict__ cb,
                                            const float* __restrict__ norms,
                                            int* __restrict__ idx_out) {
  __shared__ float xs[TILE_ROWS * XS_STRIDE];

  const int tid      = threadIdx.x;
  const int base_row = blockIdx.x * TILE_ROWS;

  // Stage 64 x 256 f32 tile of x into LDS (coalesced global, padded rows).
  for (int i = tid; i < TILE_ROWS * D_DIM; i += 128) {
    const int r = i >> 8;
    const int c = i & 255;
    xs[r * XS_STRIDE + c] = x[base_row * D_DIM + i];
  }
  __syncthreads();

  const int lane  = tid & 31;
  const int wave  = tid >> 5;
  const int nlane = lane & 15;   // A: row M%16 ; B/C: column N%16
  const int khalf = lane >> 4;   // K sub-offset 2*khalf for A and B frags

  // A fragment source: this lane's x row, offset by its K half.
  const float* arow = &xs[(wave * 16 + nlane) * XS_STRIDE + 2 * khalf];

  float bval[8];
  int   bidx[8];
#pragma unroll
  for (int r = 0; r < 8; ++r) { bval[r] = 3.4e38f; bidx[r] = 0; }

  for (int ct = 0; ct < NUM_CODES / 16; ++ct) {
    const int code = ct * 16 + nlane;              // this lane's column/code
    const float* brow = &cb[code * D_DIM + 2 * khalf];

    v8f acc = {0.f, 0.f, 0.f, 0.f, 0.f, 0.f, 0.f, 0.f};
#pragma unroll 8
    for (int kk = 0; kk < D_DIM; kk += 4) {
      v2f a = *(const v2f*)(arow + kk);            // ds_load_b64
      v2f b = *(const v2f*)(brow + kk);            // global_load_b64 (L2-hot)
      acc = __builtin_amdgcn_wmma_f32_16x16x4_f32(
          false, a, false, b, (short)0, acc, false, false);
    }

    const float nrm = norms[code];
#pragma unroll
    for (int r = 0; r < 8; ++r) {
      const float score = fmaf(-2.0f, acc[r], nrm);  // ||e||^2 - 2 z.e
      if (score < bval[r]) { bval[r] = score; bidx[r] = code; }
    }
  }

  // Min-reduce across the 16 lanes of each half (xor strides stay in-half).
#pragma unroll
  for (int r = 0; r < 8; ++r) {
    float v  = bval[r];
    int   ix = bidx[r];
#pragma unroll
    for (int off = 8; off >= 1; off >>= 1) {
      const float ov = __shfl_xor(v, off, 32);
      const int   oi = __shfl_xor(ix, off, 32);
      if (ov < v || (ov == v && oi < ix)) { v = ov; ix = oi; }
    }
    // C/D layout: VGPR r -> row r (lanes 0-15) / row r+8 (lanes 16-31)
    if (nlane == 0) idx_out[base_row + wave * 16 + khalf * 8 + r] = ix;
  }
}

// ---------------------------------------------------------------------------
// Kernel 3: gather codebook rows, write quantized_st = x + (q - x) exactly as
// the reference does, accumulate SSE and code counts. One block per row.
// ---------------------------------------------------------------------------
__global__ __launch_bounds__(256) void k_gather(const float* __restrict__ x,
                                                const float* __restrict__ cb,
                                                const int* __restrict__ idx,
                                                float* __restrict__ out,
                                                float* __restrict__ counts,
                                                float* __restrict__ sse) {
  __shared__ float red[8];
  const int row  = blockIdx.x;
  const int col  = threadIdx.x;
  const int code = idx[row];

  const float q  = cb[code * D_DIM + col];
  const float xv = x[row * D_DIM + col];
  const float d  = q - xv;
  out[row * D_DIM + col] = xv + d;   // straight-through value, fp-exact vs ref

  float s = d * d;
#pragma unroll
  for (int off = 16; off >= 1; off >>= 1) s += __shfl_xor(s, off, 32);
  if ((col & 31) == 0) red[col >> 5] = s;
  __syncthreads();
  if (col == 0) {
    float tot = 0.f;
#pragma unroll
    for (int i = 0; i < 8; ++i) tot += red[i];
    atomicAddF(sse, tot);
    atomicAddF(&counts[code], 1.0f);
  }
}

// ---------------------------------------------------------------------------
// Kernel 4: loss + perplexity. 1 block, 1024 threads (one per code).
// loss = q_loss + 0.25*e_loss = 1.25 * SSE / (N*D)   (both losses identical)
// ---------------------------------------------------------------------------
__global__ __launch_bounds__(1024) void k_final(const float* __restrict__ counts,
                                                const float* __restrict__ sse,
                                                float* __restrict__ out_tail) {
  __shared__ float red[32];
  const int t = threadIdx.x;
  const float p = counts[t] * (1.0f / (float)N_ROWS);
  float h = p * logf(p + 1e-10f);
#pragma unroll
  for (int off = 16; off >= 1; off >>= 1) h += __shfl_xor(h, off, 32);
  if ((t & 31) == 0) red[t >> 5] = h;
  __syncthreads();
  if (t == 0) {
    float H = 0.f;
#pragma unroll
    for (int i = 0; i < 32; ++i) H += red[i];
    out_tail[0] = 1.25f * sse[0] * (1.0f / (float)OUT_Q_ELEMS);
    out_tail[1] = expf(-H);
  }
}

// ---------------------------------------------------------------------------
extern "C" void kernel_launch(void* const* d_in, const int* in_sizes, int n_in,
                              void* d_out, int out_size, void* d_ws, size_t ws_size,
                              hipStream_t stream) {
  const float* x  = (const float*)d_in[0];   // [32,1024,256] f32
  const float* cb = (const float*)d_in[1];   // [1024,256] f32
  float* out = (float*)d_out;                // [8388608 quantized | loss | perp]

  float* ws     = (float*)d_ws;
  float* norms  = ws;                        // 1024 f32
  float* counts = ws + 1024;                 // 1024 f32
  float* sse    = ws + 2048;                 // 1 f32
  int*   idx    = (int*)(ws + 2304);         // 32768 i32  (~140 KB total)

  k_norms <<<NUM_CODES, 256, 0, stream>>>(cb, norms, counts, sse);
  k_vq    <<<N_ROWS / TILE_ROWS, 128, 0, stream>>>(x, cb, norms, idx);
  k_gather<<<N_ROWS, 256, 0, stream>>>(x, cb, idx, out, counts, sse);
  k_final <<<1, 1024, 0, stream>>>(counts, sse, out + OUT_Q_ELEMS);
}